// VinaFreeEnergy_7868380087052
// MI455X (gfx1250) — compile-verified
//
#include <hip/hip_runtime.h>
#include <hip/hip_bf16.h>

// Vina free energy on gfx1250 (MI455X).
// Memory-bound streaming gather: 64MB nbr_list (NT-streamed from HBM),
// 3MB X resident in L2. Wave reduction tail via V_WMMA_F32_16X16X4_F32.

typedef int   v4i __attribute__((ext_vector_type(4)));
typedef float v2f __attribute__((ext_vector_type(2)));
typedef float v8f __attribute__((ext_vector_type(8)));

#define N_ATOMS 262144
#define M_NBRS  64
#define BLOCKS  1024
#define TPB     256

__device__ __forceinline__ float pair_energy(float xi0, float xi1, float xi2,
                                             const float* __restrict__ X, int j,
                                             float w1, float w3, float w4) {
    // gather neighbor coords (L2-resident, 192MB L2 >> 3MB X)
    float xj0 = X[3 * j + 0];
    float xj1 = X[3 * j + 1];
    float xj2 = X[3 * j + 2];
    float dx = xi0 - xj0, dy = xi1 - xj1, dz = xi2 - xj2;
    float d  = fmaf(dx, dx, fmaf(dy, dy, dz * dz));      // d >= 0 always
    // repulsion (d<0) and hbond (d<0) are identically zero for d>=0 -> dropped.
    float hyd = (d < 0.5f) ? 1.0f : ((d < 1.5f) ? (1.5f - d) : 0.0f);
    float g1  = __expf(-4.0f * d * d);                   // exp(-(d/0.5)^2)
    float t   = d - 3.0f;
    float g2  = __expf(-0.25f * t * t);                  // exp(-((d-3)/2)^2)
    float e   = fmaf(w1, hyd, fmaf(w3, g1, w4 * g2));
    return (d < 8.0f) ? e : 0.0f;                        // branchless cutoff
}

__global__ void __launch_bounds__(TPB)
vina_partial(const float* __restrict__ X, const int* __restrict__ nbr,
             const float* __restrict__ combo_w, float* __restrict__ partials) {
    const int tid     = threadIdx.x;
    const int gid     = blockIdx.x * TPB + tid;
    const int gstride = gridDim.x * TPB;
    const int Q       = (N_ATOMS * M_NBRS) / 4;          // int4 elements

    // weights: uniform -> scalar loads
    const float w1 = combo_w[1];
    const float w3 = combo_w[3];
    const float w4 = combo_w[4];

    const v4i* __restrict__ nbr4 = (const v4i*)nbr;

    float acc0 = 0.0f, acc1 = 0.0f;                      // two B-matrix VGPRs
    for (int idx = gid; idx < Q; idx += gstride) {
        // prefetch next grid-stride line (speculative; OOB silently dropped)
        __builtin_prefetch(&nbr4[idx + gstride], 0, 0);
        // streamed once: non-temporal so X stays resident in L2
        v4i n = __builtin_nontemporal_load(&nbr4[idx]);
        int i = idx >> 4;                                // 16 int4 per atom row
        float xi0 = X[3 * i + 0];
        float xi1 = X[3 * i + 1];
        float xi2 = X[3 * i + 2];
        acc0 += pair_energy(xi0, xi1, xi2, X, n.x, w1, w3, w4);
        acc1 += pair_energy(xi0, xi1, xi2, X, n.y, w1, w3, w4);
        acc0 += pair_energy(xi0, xi1, xi2, X, n.z, w1, w3, w4);
        acc1 += pair_energy(xi0, xi1, xi2, X, n.w, w1, w3, w4);
    }

    // Wave-level reduction via WMMA: D = ones(16x4) x B(4x16) + 0.
    // B holds the wave's 64 partials (acc0/acc1 across 32 lanes); since A is
    // all-ones, D[m][n] = sum over the 4 K-rows of column n = acc0(n)+acc1(n)
    // +acc0(n+16)+acc1(n+16), independent of the exact K-row permutation.
    // EXEC is all-ones here (full wave, reconverged uniform control flow).
    v2f a; a[0] = 1.0f; a[1] = 1.0f;
    v2f b; b[0] = acc0; b[1] = acc1;
    v8f c = {};
    v8f dmat = __builtin_amdgcn_wmma_f32_16x16x4_f32(
        /*neg_a=*/false, a, /*neg_b=*/false, b,
        /*c_mod=*/(short)0, c, /*reuse_a=*/false, /*reuse_b=*/false);
    float v = dmat[0];                // lane l: column sum c_{l%16}
    v += __shfl_xor(v, 8, 32);        // fold 16 column sums (stays in halves)
    v += __shfl_xor(v, 4, 32);
    v += __shfl_xor(v, 2, 32);
    v += __shfl_xor(v, 1, 32);

    __shared__ float wsum[TPB / 32];
    if ((tid & 31) == 0) wsum[tid >> 5] = v;
    __syncthreads();
    if (tid == 0) {
        float s = 0.0f;
#pragma unroll
        for (int w = 0; w < TPB / 32; ++w) s += wsum[w];
        partials[blockIdx.x] = s;     // overwrite (no init needed)
    }
}

__global__ void __launch_bounds__(TPB)
vina_final(const float* __restrict__ partials,
           const float* __restrict__ nonlin_w, float* __restrict__ out) {
    __shared__ float sm[TPB];
    float s = 0.0f;
    for (int i = threadIdx.x; i < BLOCKS; i += TPB) s += partials[i];
    sm[threadIdx.x] = s;
    __syncthreads();
    for (int off = TPB / 2; off > 0; off >>= 1) {
        if (threadIdx.x < off) sm[threadIdx.x] += sm[threadIdx.x + off];
        __syncthreads();
    }
    if (threadIdx.x == 0) {
        // free_energies = thresholded / (1 + w * NROT), NROT = 1
        out[0] = sm[0] / (1.0f + nonlin_w[0]);
    }
}

extern "C" void kernel_launch(void* const* d_in, const int* in_sizes, int n_in,
                              void* d_out, int out_size, void* d_ws, size_t ws_size,
                              hipStream_t stream) {
    const float* X        = (const float*)d_in[0];
    const int*   nbr      = (const int*)d_in[1];
    const float* combo_w  = (const float*)d_in[2];
    const float* nonlin_w = (const float*)d_in[3];
    float*       out      = (float*)d_out;
    float*       partials = (float*)d_ws;   // BLOCKS floats = 4 KB scratch

    vina_partial<<<BLOCKS, TPB, 0, stream>>>(X, nbr, combo_w, partials);
    vina_final<<<1, TPB, 0, stream>>>(partials, nonlin_w, out);
}